// nonlinearLatentSDE_51711406244213
// MI455X (gfx1250) — compile-verified
//
#include <hip/hip_runtime.h>
#include <math.h>

typedef __attribute__((ext_vector_type(2))) float v2f;
typedef __attribute__((ext_vector_type(8))) float v8f;

#define DIMN 8
#define HN   100
#define HT   7          // ceil(100/16) N-tiles for GEMM1 / K-tiles for GEMM2
#define HPAD 20         // LDS row stride for the h-transpose scratch

// smem weight layout offsets (floats)
#define O_W1M 0
#define O_B1M (O_W1M + HN*DIMN)     // 800
#define O_W2M (O_B1M + HN)          // 900
#define O_B2M (O_W2M + DIMN*HN)     // 1700
#define O_W1D (O_B2M + DIMN)        // 1708
#define O_B1D (O_W1D + HN*DIMN)     // 2508
#define O_W2D (O_B1D + HN)          // 2608
#define O_B2D (O_W2D + DIMN*HN)     // 3408
#define SMEM_W (O_B2D + DIMN)       // 3416

#define LOG2E_F 1.4426950408889634f
#define LN2_F   0.6931471805599453f

// Softplus on raw v_exp_f32/v_log_f32 (no denorm guards needed):
// u = 2^(-log2e*|x|) in (0,1]; 1+u in [1,2] is never denormal, so the raw
// hardware log2 is exact-domain-safe. When u < 2^-24, 1+u rounds to 1,
// log -> 0, and softplus = max(x,0) is the correctly rounded fp32 result.
__device__ __forceinline__ float softplus_f(float x) {
    float u = __builtin_amdgcn_exp2f(-LOG2E_F * fabsf(x));
    return fmaxf(x, 0.0f) + LN2_F * __builtin_amdgcn_logf(1.0f + u);
}

// One MLP (Linear(8,100) -> softplus -> Linear(100,8)) for a 16-row tile,
// entirely on V_WMMA_F32_16X16X4_F32.
// a0/a1: A-frags of the 16x8 input tile (K=0..3, K=4..7).
// Returns C (16x16, N<8 valid) in WMMA C-layout, pre-bias.
__device__ __forceinline__ v8f mlp_tile(v2f a0, v2f a1,
                                        const float* __restrict__ sW1,
                                        const float* __restrict__ sb1,
                                        const float* __restrict__ sW2,
                                        float* hS, int l16, int half)
{
    v8f c2 = {0.f,0.f,0.f,0.f,0.f,0.f,0.f,0.f};
#pragma unroll
    for (int t = 0; t < HT; ++t) {
        // ---- GEMM1: B-frags for N-tile t (B[k][n] = W1[n*8+k], zero-padded n>=100)
        int n   = t * 16 + l16;
        int nn  = (n < HN) ? n : 0;
        float m = (n < HN) ? 1.0f : 0.0f;
        int kb  = half * 2;
        v2f b0, b1v;
        b0.x  = m * sW1[nn * DIMN + kb + 0];
        b0.y  = m * sW1[nn * DIMN + kb + 1];
        b1v.x = m * sW1[nn * DIMN + 4 + kb + 0];
        b1v.y = m * sW1[nn * DIMN + 4 + kb + 1];

        v8f acc = {0.f,0.f,0.f,0.f,0.f,0.f,0.f,0.f};
        acc = __builtin_amdgcn_wmma_f32_16x16x4_f32(false, a0, false, b0,
                                                    (short)0, acc, false, false);
        acc = __builtin_amdgcn_wmma_f32_16x16x4_f32(false, a1, false, b1v,
                                                    (short)0, acc, false, false);

        // ---- bias + softplus; scatter to LDS (C-layout -> row-major h tile)
        float bias = m * sb1[nn];
#pragma unroll
        for (int e = 0; e < 8; ++e) {
            float hv = softplus_f(acc[e] + bias);
            int M = half * 8 + e;             // row of this element
            hS[M * HPAD + l16] = hv;          // column = lane's N
        }
        asm volatile("s_wait_dscnt 0" ::: "memory");  // all 32 lanes' h visible

        // ---- GEMM2: 4 K-chunks of this tile, accumulate into c2
#pragma unroll
        for (int cc = 0; cc < 4; ++cc) {
            int kl = cc * 4 + half * 2;       // local K within tile
            v2f a2;
            a2.x = hS[l16 * HPAD + kl + 0];   // A2: lane = M, K across vgprs
            a2.y = hS[l16 * HPAD + kl + 1];

            int kg = t * 16 + kl;             // global K (H index)
            int j  = (l16 < DIMN) ? l16 : 0;  // output column
            float mj  = (l16 < DIMN) ? 1.0f : 0.0f;
            float m0  = (kg + 0 < HN) ? mj : 0.0f;
            float m1  = (kg + 1 < HN) ? mj : 0.0f;
            int k0 = (kg + 0 < HN) ? kg + 0 : 0;
            int k1 = (kg + 1 < HN) ? kg + 1 : 0;
            v2f b2;
            b2.x = m0 * sW2[j * HN + k0];     // B2[k][j] = W2[j*100+k]
            b2.y = m1 * sW2[j * HN + k1];

            c2 = __builtin_amdgcn_wmma_f32_16x16x4_f32(false, a2, false, b2,
                                                       (short)0, c2, false, false);
        }
    }
    return c2;
}

__global__ void sde_zero_loss(float* p) { if (threadIdx.x == 0) p[0] = 0.0f; }

__global__ __launch_bounds__(256)
void sde_nll_kernel(const float* __restrict__ zt1, const float* __restrict__ zt2,
                    const float* __restrict__ dt,
                    const float* __restrict__ W1m, const float* __restrict__ b1m,
                    const float* __restrict__ W2m, const float* __restrict__ b2m,
                    const float* __restrict__ W1d, const float* __restrict__ b1d,
                    const float* __restrict__ W2d, const float* __restrict__ b2d,
                    float* __restrict__ loss_out, float* __restrict__ mu_out,
                    float* __restrict__ sig_out, int nrows)
{
    __shared__ float smem[SMEM_W];
    __shared__ float sH[8][16 * HPAD];   // per-wave transpose scratch

    // ---- stage weights into LDS
    for (int i = threadIdx.x; i < HN * DIMN; i += 256) {
        smem[O_W1M + i] = W1m[i];
        smem[O_W1D + i] = W1d[i];
        smem[O_W2M + i] = W2m[i];
        smem[O_W2D + i] = W2d[i];
    }
    for (int i = threadIdx.x; i < HN; i += 256) {
        smem[O_B1M + i] = b1m[i];
        smem[O_B1D + i] = b1d[i];
    }
    if (threadIdx.x < DIMN) {
        smem[O_B2M + threadIdx.x] = b2m[threadIdx.x];
        smem[O_B2D + threadIdx.x] = b2d[threadIdx.x];
    }
    __syncthreads();

    const int lane   = threadIdx.x & 31;
    const int l16    = lane & 15;
    const int half   = (lane >> 4) & 1;
    const int waveId = threadIdx.x >> 5;
    float* hS = &sH[waveId][0];

    const int wgid   = (blockIdx.x * blockDim.x + threadIdx.x) >> 5;
    const int nwaves = (gridDim.x * blockDim.x) >> 5;

    const float dt0  = dt[0];
    const float sqdt = sqrtf(dt0);
    const float C0   = 4.0f * 1.8378770664093453f;   // (DIM/2)*log(2*pi)

    const int ntiles = nrows >> 4;
    for (int tile = wgid; tile < ntiles; tile += nwaves) {
        const int row0 = tile << 4;

        // A1 frags of zt1 tile: lane = row M, K = c*4 + half*2 + i
        const float* z1p = zt1 + (size_t)(row0 + l16) * DIMN;
        v2f a0, a1;
        a0.x = z1p[half * 2 + 0];
        a0.y = z1p[half * 2 + 1];
        a1.x = z1p[4 + half * 2 + 0];
        a1.y = z1p[4 + half * 2 + 1];

        v8f cm = mlp_tile(a0, a1, smem + O_W1M, smem + O_B1M, smem + O_W2M,
                          hS, l16, half);
        v8f cd = mlp_tile(a0, a1, smem + O_W1D, smem + O_B1D, smem + O_W2D,
                          hS, l16, half);

        // ---- epilogue: per-lane column j = l16 (valid j<8), rows M = half*8+e
        const int  j  = l16;
        const bool jv = (j < DIMN);
        const float b2mj = jv ? smem[O_B2M + j] : 0.0f;
        const float b2dj = jv ? smem[O_B2D + j] : 0.0f;

        float part[8];
#pragma unroll
        for (int e = 0; e < 8; ++e) {
            const int r  = row0 + half * 8 + e;
            const float mu = (cm[e] + b2mj) * dt0;
            const float g  = cd[e] + b2dj;           // log D
            const float Dv = __builtin_amdgcn_exp2f(LOG2E_F * g);
            if (jv) {
                mu_out[(size_t)r * DIMN + j]  = mu;
                sig_out[(size_t)r * DIMN + j] = Dv * sqdt;
            }
            float dz = 0.0f;
            if (jv) dz = zt2[(size_t)r * DIMN + j] - zt1[(size_t)r * DIMN + j];
            const float dm = dz - mu;
            const float iv = __builtin_amdgcn_exp2f(-2.0f * LOG2E_F * g); // exp(-2g)
            // t1 contribution 2g; (t2+t3+t4) = (dz-mu)^2 * exp(-2g)
            part[e] = jv ? (2.0f * g + dm * dm * iv) : 0.0f;
        }

        // reduce over the 8 valid column-lanes (xor within 8-lane groups)
#pragma unroll
        for (int e = 0; e < 8; ++e) {
            float s = part[e];
            s += __shfl_xor(s, 1, 32);
            s += __shfl_xor(s, 2, 32);
            s += __shfl_xor(s, 4, 32);
            part[e] = s;
        }
        float local = 0.0f;
#pragma unroll
        for (int e = 0; e < 8; ++e) local += 0.5f * part[e] + C0;
        const float tot = local + __shfl_xor(local, 16, 32); // rows 0-7 + rows 8-15
        if (lane == 0) atomicAdd(loss_out, tot);
    }
}

extern "C" void kernel_launch(void* const* d_in, const int* in_sizes, int n_in,
                              void* d_out, int out_size, void* d_ws, size_t ws_size,
                              hipStream_t stream)
{
    const float* zt1 = (const float*)d_in[0];
    const float* zt2 = (const float*)d_in[1];
    const float* dt  = (const float*)d_in[2];
    const float* W1m = (const float*)d_in[3];
    const float* b1m = (const float*)d_in[4];
    const float* W2m = (const float*)d_in[5];
    const float* b2m = (const float*)d_in[6];
    const float* W1d = (const float*)d_in[7];
    const float* b1d = (const float*)d_in[8];
    const float* W2d = (const float*)d_in[9];
    const float* b2d = (const float*)d_in[10];

    const int nrows = in_sizes[0] / DIMN;      // B = 262144

    float* loss = (float*)d_out;
    float* mu   = loss + 1;
    float* sig  = mu + (size_t)nrows * DIMN;

    sde_zero_loss<<<1, 32, 0, stream>>>(loss);
    sde_nll_kernel<<<256, 256, 0, stream>>>(zt1, zt2, dt,
                                            W1m, b1m, W2m, b2m,
                                            W1d, b1d, W2d, b2d,
                                            loss, mu, sig, nrows);
}